// CriticNetwork_14181982012211
// MI455X (gfx1250) — compile-verified
//
#include <hip/hip_runtime.h>

#define N_NODES 50000
#define N_EDGES 800000
#define E_TOT   (N_EDGES + N_NODES)   // 850000 incl. self-loops
#define N_GRAPHS 64
#define NEG_SLOPE 0.2f

typedef float v2f __attribute__((ext_vector_type(2)));
typedef float v8f __attribute__((ext_vector_type(8)));

// ---------------------------------------------------------------------------
// GEMM: Out[M, HC] = A[M, K] @ W[K, HC], M = N_NODES (multiple of 16).
// One wave32 computes a 16x16 tile via V_WMMA_F32_16X16X4_F32, K stepped by 4.
// A frag (16x4 f32): lanes 0-15 hold rows 0-15 K=k0..k0+1; lanes 16-31 K=k0+2..3.
// C/D frag: VGPR i -> row (i + 8*(lane>=16)), col = lane&15.
// ---------------------------------------------------------------------------
template<int K, int HC>
__global__ void gemm_wmma_f32(const float* __restrict__ A,
                              const float* __restrict__ W,
                              float* __restrict__ Out, int tilesTotal) {
  int wave = (int)((blockIdx.x * blockDim.x + threadIdx.x) >> 5);
  if (wave >= tilesTotal) return;      // whole-wave uniform exit: EXEC all-1 for WMMA
  const int TN = HC / 16;
  int tm = wave / TN;
  int tn = wave - tm * TN;
  int lane = threadIdx.x & 31;
  int half = lane >> 4;                // 0: K pair (k0,k0+1), 1: (k0+2,k0+3)
  int l15  = lane & 15;
  const float* arow = A + (size_t)(tm * 16 + l15) * K;
  int col = tn * 16 + l15;
  v8f acc = {};
#pragma unroll
  for (int k0 = 0; k0 < K; k0 += 4) {
    int ka = k0 + half * 2;
    v2f a = *(const v2f*)(arow + ka);            // 8B aligned global_load_b64
    v2f b;
    b.x = W[(size_t)ka * HC + col];
    b.y = W[(size_t)(ka + 1) * HC + col];
    acc = __builtin_amdgcn_wmma_f32_16x16x4_f32(
        /*neg_a=*/false, a, /*neg_b=*/false, b,
        /*c_mod=*/(short)0, acc, /*reuse_a=*/false, /*reuse_b=*/false);
  }
#pragma unroll
  for (int i = 0; i < 8; ++i)
    Out[(size_t)(tm * 16 + half * 8 + i) * HC + col] = acc[i];
}

// ---------------------------------------------------------------------------
// Per-(node, head) attention scalars: a = sum_c h[n,h,c] * att[h,c]
// ---------------------------------------------------------------------------
template<int H, int C, int HC>
__global__ void att_scores_k(const float* __restrict__ h,
                             const float* __restrict__ att_s,
                             const float* __restrict__ att_d,
                             float* __restrict__ asrc, float* __restrict__ adst) {
  int t = blockIdx.x * blockDim.x + threadIdx.x;
  if (t >= N_NODES * H) return;
  int n  = t / H;
  int hh = t - n * H;
  const float* hp = h + (size_t)n * HC + hh * C;
  const float* as = att_s + hh * C;
  const float* ad = att_d + hh * C;
  float s = 0.f, d = 0.f;
#pragma unroll
  for (int c = 0; c < C; ++c) { float v = hp[c]; s += v * as[c]; d += v * ad[c]; }
  asrc[t] = s;
  adst[t] = d;
}

// order-preserving float <-> uint mapping for atomic max on signed floats
__device__ __forceinline__ unsigned f2ord(float f) {
  unsigned u = __float_as_uint(f);
  return (u & 0x80000000u) ? ~u : (u | 0x80000000u);
}
__device__ __forceinline__ float ord2f(unsigned k) {
  return __uint_as_float((k & 0x80000000u) ? (k ^ 0x80000000u) : ~k);
}

__device__ __forceinline__ void edge_nodes(int e, int& s, int& d, const int* ei) {
  if (e < N_EDGES) { s = ei[e]; d = ei[N_EDGES + e]; }
  else             { s = d = e - N_EDGES; }            // self loop
}

template<int H>
__global__ void edge_max_k(const int* __restrict__ ei, const float* __restrict__ asrc,
                           const float* __restrict__ adst, unsigned* __restrict__ mkey) {
  int e = blockIdx.x * blockDim.x + threadIdx.x;
  if (e >= E_TOT) return;
  int s, d; edge_nodes(e, s, d, ei);
#pragma unroll
  for (int h = 0; h < H; ++h) {
    float v = asrc[s * H + h] + adst[d * H + h];
    v = v < 0.f ? v * NEG_SLOPE : v;
    atomicMax(&mkey[d * H + h], f2ord(v));
  }
}

template<int H>
__global__ void edge_exp_k(const int* __restrict__ ei, const float* __restrict__ asrc,
                           const float* __restrict__ adst, const unsigned* __restrict__ mkey,
                           float* __restrict__ exbuf, float* __restrict__ denom) {
  int e = blockIdx.x * blockDim.x + threadIdx.x;
  if (e >= E_TOT) return;
  int s, d; edge_nodes(e, s, d, ei);
#pragma unroll
  for (int h = 0; h < H; ++h) {
    float v = asrc[s * H + h] + adst[d * H + h];
    v = v < 0.f ? v * NEG_SLOPE : v;
    float ex = __expf(v - ord2f(mkey[d * H + h]));
    exbuf[(size_t)e * H + h] = ex;
    atomicAdd(&denom[d * H + h], ex);
  }
}

// HC/4 threads per edge; each scatters 4 channels with fp32 atomics
template<int H, int C, int HC>
__global__ void edge_agg_k(const int* __restrict__ ei, const float* __restrict__ hfeat,
                           const float* __restrict__ exbuf, const float* __restrict__ denom,
                           float* __restrict__ outb) {
  const int TPE = HC / 4;
  long long t = (long long)blockIdx.x * blockDim.x + threadIdx.x;
  long long e = t / TPE;
  if (e >= E_TOT) return;
  int c0 = (int)(t - e * TPE) * 4;
  int s, d; edge_nodes((int)e, s, d, ei);
  int hh = c0 / C;
  float alpha = exbuf[e * H + hh] / (denom[d * H + hh] + 1e-16f);
  const float4 hv = *(const float4*)(hfeat + (size_t)s * HC + c0);
  float* ob = outb + (size_t)d * HC + c0;
  atomicAdd(ob + 0, hv.x * alpha);
  atomicAdd(ob + 1, hv.y * alpha);
  atomicAdd(ob + 2, hv.z * alpha);
  atomicAdd(ob + 3, hv.w * alpha);
}

__global__ void bias_act_k(const float* __restrict__ in, const float* __restrict__ b,
                           float* __restrict__ out, int total, int HC, int do_relu) {
  int t = blockIdx.x * blockDim.x + threadIdx.x;
  if (t >= total) return;
  float v = in[t] + b[t % HC];
  if (do_relu) v = fmaxf(v, 0.f);
  out[t] = v;
}

__global__ void pool_k(const float* __restrict__ h3, const int* __restrict__ batch,
                       float* __restrict__ pooled, float* __restrict__ cnt) {
  int n = blockIdx.x * blockDim.x + threadIdx.x;
  if (n >= N_NODES) return;
  int g = batch[n];
#pragma unroll
  for (int c = 0; c < 32; c += 4) {
    float4 v = *(const float4*)(h3 + (size_t)n * 32 + c);
    atomicAdd(&pooled[g * 32 + c + 0], v.x);
    atomicAdd(&pooled[g * 32 + c + 1], v.y);
    atomicAdd(&pooled[g * 32 + c + 2], v.z);
    atomicAdd(&pooled[g * 32 + c + 3], v.w);
  }
  atomicAdd(&cnt[g], 1.0f);
}

__global__ void mlp_k(const float* __restrict__ pooled, const float* __restrict__ cnt,
                      const float* __restrict__ actions,
                      const float* __restrict__ lw1, const float* __restrict__ lb1,
                      const float* __restrict__ lw2, const float* __restrict__ lb2,
                      const float* __restrict__ lw3, const float* __restrict__ lb3,
                      float* __restrict__ out) {
  int g = threadIdx.x;
  if (g >= N_GRAPHS) return;
  float sa[36], t1[32], t2[32];
  float ic = 1.0f / fmaxf(cnt[g], 1.0f);
#pragma unroll
  for (int c = 0; c < 32; ++c) sa[c] = pooled[g * 32 + c] * ic;
#pragma unroll
  for (int a = 0; a < 4; ++a) sa[32 + a] = actions[g * 4 + a];
#pragma unroll
  for (int j = 0; j < 32; ++j) {
    float s = lb1[j];
    for (int i = 0; i < 36; ++i) s += sa[i] * lw1[i * 32 + j];
    t1[j] = s;
  }
#pragma unroll
  for (int j = 0; j < 32; ++j) {
    float s = lb2[j];
    for (int i = 0; i < 32; ++i) s += t1[i] * lw2[i * 32 + j];
    t2[j] = s;
  }
  float s = lb3[0];
#pragma unroll
  for (int i = 0; i < 32; ++i) s += t2[i] * lw3[i];
  out[g] = s;
}

// ---------------------------------------------------------------------------
extern "C" void kernel_launch(void* const* d_in, const int* in_sizes, int n_in,
                              void* d_out, int out_size, void* d_ws, size_t ws_size,
                              hipStream_t stream) {
  (void)in_sizes; (void)n_in; (void)out_size; (void)ws_size;
  const float* x       = (const float*)d_in[0];
  const int*   ei      = (const int*)d_in[1];
  const int*   batch   = (const int*)d_in[2];
  const float* actions = (const float*)d_in[3];
  const float* W1 = (const float*)d_in[4];
  const float* as1 = (const float*)d_in[5];
  const float* ad1 = (const float*)d_in[6];
  const float* b1 = (const float*)d_in[7];
  const float* W2 = (const float*)d_in[8];
  const float* as2 = (const float*)d_in[9];
  const float* ad2 = (const float*)d_in[10];
  const float* b2 = (const float*)d_in[11];
  const float* W3 = (const float*)d_in[12];
  const float* as3 = (const float*)d_in[13];
  const float* ad3 = (const float*)d_in[14];
  const float* b3 = (const float*)d_in[15];
  const float* lw1 = (const float*)d_in[16];
  const float* lb1 = (const float*)d_in[17];
  const float* lw2 = (const float*)d_in[18];
  const float* lb2 = (const float*)d_in[19];
  const float* lw3 = (const float*)d_in[20];
  const float* lb3 = (const float*)d_in[21];
  float* out = (float*)d_out;

  // carve scratch
  char* base = (char*)d_ws;
  size_t off = 0;
  auto carve = [&](size_t bytes) -> char* {
    char* p = base + off;
    off = (off + bytes + 255) & ~(size_t)255;
    return p;
  };
  float*    hfeat  = (float*)carve((size_t)N_NODES * 128 * 4);  // gemm output
  float*    hbuf   = (float*)carve((size_t)N_NODES * 128 * 4);  // agg / activations
  float*    asrc   = (float*)carve((size_t)N_NODES * 4 * 4);
  float*    adst   = (float*)carve((size_t)N_NODES * 4 * 4);
  unsigned* mkey   = (unsigned*)carve((size_t)N_NODES * 4 * 4);
  float*    denom  = (float*)carve((size_t)N_NODES * 4 * 4);
  float*    exbuf  = (float*)carve((size_t)E_TOT * 4 * 4);
  float*    pooled = (float*)carve((size_t)N_GRAPHS * 32 * 4);
  float*    cnt    = (float*)carve((size_t)N_GRAPHS * 4);

  const int BLK = 256;
  const unsigned eb  = (E_TOT + BLK - 1) / BLK;

  // ---------------- Layer 1: K=32, H=4, C=32, HC=128 ----------------
  {
    int tiles = (N_NODES / 16) * (128 / 16);
    gemm_wmma_f32<32, 128><<<(tiles + 7) / 8, BLK, 0, stream>>>(x, W1, hfeat, tiles);
    hipMemsetAsync(hbuf, 0, (size_t)N_NODES * 128 * 4, stream);
    hipMemsetAsync(mkey, 0, (size_t)N_NODES * 4 * 4, stream);
    hipMemsetAsync(denom, 0, (size_t)N_NODES * 4 * 4, stream);
    att_scores_k<4, 32, 128><<<(N_NODES * 4 + BLK - 1) / BLK, BLK, 0, stream>>>(hfeat, as1, ad1, asrc, adst);
    edge_max_k<4><<<eb, BLK, 0, stream>>>(ei, asrc, adst, mkey);
    edge_exp_k<4><<<eb, BLK, 0, stream>>>(ei, asrc, adst, mkey, exbuf, denom);
    unsigned ab = (unsigned)(((size_t)E_TOT * 32 + BLK - 1) / BLK);
    edge_agg_k<4, 32, 128><<<ab, BLK, 0, stream>>>(ei, hfeat, exbuf, denom, hbuf);
    bias_act_k<<<((size_t)N_NODES * 128 + BLK - 1) / BLK, BLK, 0, stream>>>(hbuf, b1, hbuf, N_NODES * 128, 128, 1);
  }
  // ---------------- Layer 2: K=128, H=4, C=32, HC=128 ----------------
  {
    int tiles = (N_NODES / 16) * (128 / 16);
    gemm_wmma_f32<128, 128><<<(tiles + 7) / 8, BLK, 0, stream>>>(hbuf, W2, hfeat, tiles);
    hipMemsetAsync(hbuf, 0, (size_t)N_NODES * 128 * 4, stream);   // stream-ordered after gemm read
    hipMemsetAsync(mkey, 0, (size_t)N_NODES * 4 * 4, stream);
    hipMemsetAsync(denom, 0, (size_t)N_NODES * 4 * 4, stream);
    att_scores_k<4, 32, 128><<<(N_NODES * 4 + BLK - 1) / BLK, BLK, 0, stream>>>(hfeat, as2, ad2, asrc, adst);
    edge_max_k<4><<<eb, BLK, 0, stream>>>(ei, asrc, adst, mkey);
    edge_exp_k<4><<<eb, BLK, 0, stream>>>(ei, asrc, adst, mkey, exbuf, denom);
    unsigned ab = (unsigned)(((size_t)E_TOT * 32 + BLK - 1) / BLK);
    edge_agg_k<4, 32, 128><<<ab, BLK, 0, stream>>>(ei, hfeat, exbuf, denom, hbuf);
    bias_act_k<<<((size_t)N_NODES * 128 + BLK - 1) / BLK, BLK, 0, stream>>>(hbuf, b2, hbuf, N_NODES * 128, 128, 1);
  }
  // ---------------- Layer 3: K=128, H=1, C=32, HC=32 (mean over 1 head = id) ----
  {
    int tiles = (N_NODES / 16) * (32 / 16);
    gemm_wmma_f32<128, 32><<<(tiles + 7) / 8, BLK, 0, stream>>>(hbuf, W3, hfeat, tiles);
    hipMemsetAsync(hbuf, 0, (size_t)N_NODES * 32 * 4, stream);
    hipMemsetAsync(mkey, 0, (size_t)N_NODES * 4, stream);
    hipMemsetAsync(denom, 0, (size_t)N_NODES * 4, stream);
    att_scores_k<1, 32, 32><<<(N_NODES + BLK - 1) / BLK, BLK, 0, stream>>>(hfeat, as3, ad3, asrc, adst);
    edge_max_k<1><<<eb, BLK, 0, stream>>>(ei, asrc, adst, mkey);
    edge_exp_k<1><<<eb, BLK, 0, stream>>>(ei, asrc, adst, mkey, exbuf, denom);
    unsigned ab = (unsigned)(((size_t)E_TOT * 8 + BLK - 1) / BLK);
    edge_agg_k<1, 32, 32><<<ab, BLK, 0, stream>>>(ei, hfeat, exbuf, denom, hbuf);
    bias_act_k<<<((size_t)N_NODES * 32 + BLK - 1) / BLK, BLK, 0, stream>>>(hbuf, b3, hbuf, N_NODES * 32, 32, 0);
  }
  // ---------------- Pool + MLP head ----------------
  hipMemsetAsync(pooled, 0, (size_t)N_GRAPHS * 32 * 4, stream);
  hipMemsetAsync(cnt, 0, (size_t)N_GRAPHS * 4, stream);
  pool_k<<<(N_NODES + BLK - 1) / BLK, BLK, 0, stream>>>(hbuf, batch, pooled, cnt);
  mlp_k<<<1, 64, 0, stream>>>(pooled, cnt, actions, lw1, lb1, lw2, lb2, lw3, lb3, out);
}